// Graph_55259049230375
// MI455X (gfx1250) — compile-verified
//
#include <hip/hip_runtime.h>
#include <math.h>

// MI455X / gfx1250, wave32. fp32 matrix path via V_WMMA_F32_16X16X4_F32.
//
// Problem: B=2048, T=200, D=128 session attention + STAMP head, all fp32.
// ~96 GFLOP fp32 vs ~2.5 GB workspace traffic (~110us @ 23.3 TB/s HBM):
// compute-dominated -> use fp32 WMMA (keeps reference numerics, uses matrix pipes).
//
// Workspace layout (floats): Q | K | V, each B*208*128. dec aliases Q
// (safe: each wave fully loads its own Q rows before writing the same dec rows;
// rows are wave-private). Requires ws_size >= 3*2048*208*128*4 B (~654 MB).

typedef __attribute__((ext_vector_type(2))) float v2f;
typedef __attribute__((ext_vector_type(8))) float v8f;

#define B_      2048
#define T_      200
#define D_      128
#define NNODE_  200
#define TP      208          // T padded to 13 tiles of 16
#define NT      13
#define HSTR    132          // LDS row stride (floats): 132%64=4 -> conflict-free A reads
#define SSTR    210          // score scratch row stride: 210%64=18 -> conflict-free
#define PBS     (TP * D_)    // 26624 floats per per-batch matrix
#define NEGC    (-4294967295.0f)

__device__ __forceinline__ v8f wmma4(v2f a, v2f b, v8f c) {
  // D = A(16x4 f32) * B(4x16 f32) + C(16x16 f32)
  return __builtin_amdgcn_wmma_f32_16x16x4_f32(false, a, false, b, (short)0, c,
                                               false, false);
}

// ---------------------------------------------------------------- kernel 1
// Per b: gather seq_h -> LDS, Q/K/V = relu(H @ W*) via WMMA, store to ws.
__global__ void qkv_kernel(const float* __restrict__ re, const int* __restrict__ alias,
                           const float* __restrict__ Wq, const float* __restrict__ Wk,
                           const float* __restrict__ Wv,
                           float* __restrict__ wsQ, float* __restrict__ wsK,
                           float* __restrict__ wsV) {
  extern __shared__ float smem[];
  float* sH = smem;                 // TP*HSTR
  float* sW = smem + TP * HSTR;     // 3*D*D (Wq,Wk,Wv)
  const int b = blockIdx.x, tid = threadIdx.x;

  for (int idx = tid; idx < TP * D_; idx += 256) {
    int t = idx >> 7, d = idx & 127;
    float v = 0.f;
    if (t < T_) {
      int a = alias[b * T_ + t];
      v = re[((size_t)b * NNODE_ + a) * D_ + d];
    }
    sH[t * HSTR + d] = v;
  }
  for (int idx = tid; idx < 3 * D_ * D_; idx += 256) {
    int mat = idx >> 14, r = idx & 16383;
    const float* src = (mat == 0) ? Wq : (mat == 1) ? Wk : Wv;
    sW[idx] = src[r];
  }
  __syncthreads();

  const int wave = tid >> 5, lane = tid & 31, lh = lane & 15, hi = lane >> 4;
  for (int tile = wave; tile < 3 * NT * 8; tile += 8) {
    int mat = tile / (NT * 8), rem = tile % (NT * 8);
    int i = rem >> 3, n0 = (rem & 7) << 4;
    const float* W = sW + mat * D_ * D_;
    const float* arow = sH + (i * 16 + lh) * HSTR;
    v8f acc = {0.f, 0.f, 0.f, 0.f, 0.f, 0.f, 0.f, 0.f};
#pragma unroll
    for (int c = 0; c < 32; ++c) {
      int k = 4 * c + 2 * hi;                       // A/B lane layout, ISA 7.12.2
      v2f a = *(const v2f*)(arow + k);
      v2f bb;
      bb.x = W[k * D_ + n0 + lh];
      bb.y = W[(k + 1) * D_ + n0 + lh];
      acc = wmma4(a, bb, acc);
    }
    float* dst = ((mat == 0) ? wsQ : (mat == 1) ? wsK : wsV) + (size_t)b * PBS;
#pragma unroll
    for (int r = 0; r < 8; ++r) {                   // C/D layout: M = r + 8*hi
      int row = i * 16 + r + 8 * hi;
      float v = acc[r];
      dst[row * D_ + n0 + lh] = v > 0.f ? v : 0.f;  // relu
    }
  }
}

// ---------------------------------------------------------------- kernel 2
// Per b: K,V in LDS; per query tile: S=Q*K^T (WMMA), mask+softmax via LDS
// scratch, dec = attn*V + seq_h (residual regathered), store dec (aliases Q).
__global__ void attn_kernel(const float* __restrict__ re, const int* __restrict__ alias,
                            const int* __restrict__ seqlen,
                            const float* __restrict__ wsQ, const float* __restrict__ wsK,
                            const float* __restrict__ wsV, float* __restrict__ wsDec) {
  extern __shared__ float smem[];
  float* sK = smem;                       // TP*HSTR
  float* sV = smem + TP * HSTR;           // TP*HSTR
  float* sS0 = smem + 2 * TP * HSTR;      // 8 waves * 16*SSTR
  const int b = blockIdx.x, tid = threadIdx.x;
  const int len = seqlen[b];
  const float* Kb = wsK + (size_t)b * PBS;
  const float* Vb = wsV + (size_t)b * PBS;
  for (int idx = tid; idx < TP * D_; idx += 256) {
    int t = idx >> 7, d = idx & 127;
    sK[t * HSTR + d] = Kb[idx];
    sV[t * HSTR + d] = Vb[idx];
  }
  __syncthreads();

  const int wave = tid >> 5, lane = tid & 31, lh = lane & 15, hi = lane >> 4;
  float* sS = sS0 + wave * (16 * SSTR);
  const float scale = 0.088388347648318447f;  // 1/sqrt(128)

  for (int pass = 0; pass < 2; ++pass) {      // uniform barriers across waves
    const int qt = wave + 8 * pass;
    const bool active = (qt < NT);
    if (active) {
      v2f qa[32];
      const float* Qrow = wsQ + (size_t)b * PBS + (qt * 16 + lh) * D_;
#pragma unroll
      for (int c = 0; c < 32; ++c) qa[c] = *(const v2f*)(Qrow + 4 * c + 2 * hi);
      for (int j = 0; j < NT; ++j) {
        v8f acc = {0.f, 0.f, 0.f, 0.f, 0.f, 0.f, 0.f, 0.f};
#pragma unroll
        for (int c = 0; c < 32; ++c) {
          // B[k=d][n=s] = K[s][d]: K-contiguous row read -> ds_load_b64
          v2f bb = *(const v2f*)(sK + (j * 16 + lh) * HSTR + 4 * c + 2 * hi);
          acc = wmma4(qa[c], bb, acc);
        }
#pragma unroll
        for (int r = 0; r < 8; ++r) {
          int m = r + 8 * hi, s = j * 16 + lh;
          sS[m * SSTR + s] = (s < len) ? acc[r] * scale : NEGC;  // key mask
        }
      }
    }
    __syncthreads();
    if (active) {  // softmax: lane owns row lh, half-columns selected by hi
      float* rowp = sS + lh * SSTR + hi * 104;
      float mx = -3.0e38f;
      for (int s = 0; s < 104; ++s) mx = fmaxf(mx, rowp[s]);
      mx = fmaxf(mx, __shfl_xor(mx, 16, 32));
      float sum = 0.f;
      for (int s = 0; s < 104; ++s) { float e = __expf(rowp[s] - mx); rowp[s] = e; sum += e; }
      sum += __shfl_xor(sum, 16, 32);
      float qm = ((qt * 16 + lh) < len) ? 1.f : 0.f;  // query mask
      float inv = qm / sum;
      for (int s = 0; s < 104; ++s) rowp[s] *= inv;
    }
    __syncthreads();
    if (active) {
      float* Db = wsDec + (size_t)b * PBS;
      for (int nt = 0; nt < 8; ++nt) {
        int n0 = nt * 16;
        v8f acc;
#pragma unroll
        for (int r = 0; r < 8; ++r) {          // residual: seq_h regather
          int t = qt * 16 + r + 8 * hi;
          float h = 0.f;
          if (t < T_) {
            int a = alias[b * T_ + t];
            h = re[((size_t)b * NNODE_ + a) * D_ + n0 + lh];
          }
          acc[r] = h;
        }
#pragma unroll
        for (int c = 0; c < 52; ++c) {         // K-dim = 208 over s
          int s = 4 * c + 2 * hi;
          v2f a = *(const v2f*)(sS + lh * SSTR + s);
          v2f bb;
          bb.x = sV[s * HSTR + n0 + lh];
          bb.y = sV[(s + 1) * HSTR + n0 + lh];
          acc = wmma4(a, bb, acc);
        }
#pragma unroll
        for (int r = 0; r < 8; ++r) {
          int t = qt * 16 + r + 8 * hi;
          Db[t * D_ + n0 + lh] = acc[r];
        }
      }
    }
    __syncthreads();
  }
}

// ---------------------------------------------------------------- kernel 3
// STAMP head: last_p = dec[last] @ u_w1; m = sigmoid(last_p + dec@u_w2 + u_b);
// coef = (m.u_v)*mask; out = [sum_t coef*dec , last_p].
__global__ void stamp_kernel(const int* __restrict__ seqlen, const float* __restrict__ wsDec,
                             const float* __restrict__ u_w1, const float* __restrict__ u_w2,
                             const float* __restrict__ u_b, const float* __restrict__ u_v,
                             float* __restrict__ out) {
  extern __shared__ float smem[];
  float* sW2 = smem;              // D*D
  float* sLp = smem + D_ * D_;    // 128
  float* sUb = sLp + 128;         // 128
  float* sUv = sUb + 128;         // 128
  float* sCoef = sUv + 128;       // TP
  const int b = blockIdx.x, tid = threadIdx.x;
  const int len = seqlen[b];
  const float* Db = wsDec + (size_t)b * PBS;

  for (int idx = tid; idx < D_ * D_; idx += 256) sW2[idx] = u_w2[idx];
  if (tid < D_) { sUb[tid] = u_b[tid]; sUv[tid] = u_v[tid]; }
  if (tid < D_) {
    const float* lrow = Db + (len - 1) * D_;     // last valid position
    float acc = 0.f;
    for (int d = 0; d < D_; ++d) acc += lrow[d] * u_w1[d * D_ + tid];
    sLp[tid] = acc;
    out[(size_t)b * 2 * D_ + D_ + tid] = acc;    // second half of output
  }
  __syncthreads();

  const int wave = tid >> 5, lane = tid & 31, lh = lane & 15, hi = lane >> 4;
  for (int pass = 0; pass < 2; ++pass) {
    const int qt = wave + 8 * pass;
    if (qt < NT) {
      v2f da[32];
      const float* drow = Db + (qt * 16 + lh) * D_;
#pragma unroll
      for (int c = 0; c < 32; ++c) da[c] = *(const v2f*)(drow + 4 * c + 2 * hi);
      float cpart[8];
#pragma unroll
      for (int r = 0; r < 8; ++r) cpart[r] = 0.f;
      for (int nt = 0; nt < 8; ++nt) {
        int n0 = nt * 16;
        v8f acc = {0.f, 0.f, 0.f, 0.f, 0.f, 0.f, 0.f, 0.f};
#pragma unroll
        for (int c = 0; c < 32; ++c) {
          int k = 4 * c + 2 * hi;
          v2f bb;
          bb.x = sW2[k * D_ + n0 + lh];
          bb.y = sW2[(k + 1) * D_ + n0 + lh];
          acc = wmma4(da[c], bb, acc);
        }
        int col = n0 + lh;
        float add = sLp[col] + sUb[col];
        float uvv = sUv[col];
#pragma unroll
        for (int r = 0; r < 8; ++r) {
          float x = acc[r] + add;
          float sg = 1.f / (1.f + __expf(-x));   // sigmoid
          cpart[r] += sg * uvv;                  // partial m @ u_v^T
        }
      }
#pragma unroll
      for (int r = 0; r < 8; ++r) {              // reduce over N (16 lanes/half)
        float v = cpart[r];
        v += __shfl_xor(v, 1, 32);
        v += __shfl_xor(v, 2, 32);
        v += __shfl_xor(v, 4, 32);
        v += __shfl_xor(v, 8, 32);
        if (lh == 0) {
          int t = qt * 16 + r + 8 * hi;
          sCoef[t] = (t < len) ? v : 0.f;        // mask
        }
      }
    }
  }
  __syncthreads();
  if (tid < D_) {
    float acc = 0.f;
    for (int t = 0; t < TP; ++t) acc += sCoef[t] * Db[t * D_ + tid];
    out[(size_t)b * 2 * D_ + tid] = acc;
  }
}

// ---------------------------------------------------------------- launcher
extern "C" void kernel_launch(void* const* d_in, const int* in_sizes, int n_in,
                              void* d_out, int out_size, void* d_ws, size_t ws_size,
                              hipStream_t stream) {
  (void)in_sizes; (void)n_in; (void)out_size; (void)ws_size;
  const float* re   = (const float*)d_in[0];
  const int*   alia = (const int*)d_in[1];
  const int*   slen = (const int*)d_in[2];
  const float* Wq   = (const float*)d_in[3];
  const float* Wk   = (const float*)d_in[4];
  const float* Wv   = (const float*)d_in[5];
  const float* u_w1 = (const float*)d_in[6];
  const float* u_w2 = (const float*)d_in[7];
  const float* u_b  = (const float*)d_in[8];
  const float* u_v  = (const float*)d_in[9];
  float* out = (float*)d_out;

  float* ws   = (float*)d_ws;
  float* wsQ  = ws;                             // later reused as dec
  float* wsK  = ws + (size_t)B_ * PBS;
  float* wsV  = ws + 2 * (size_t)B_ * PBS;
  float* wsDec = wsQ;

  size_t sm1 = (size_t)(TP * HSTR + 3 * D_ * D_) * sizeof(float);        // 306,432 B
  size_t sm2 = (size_t)(2 * TP * HSTR + 8 * 16 * SSTR) * sizeof(float);  // 327,168 B
  size_t sm3 = (size_t)(D_ * D_ + 3 * 128 + TP) * sizeof(float);         //  67,904 B
  hipFuncSetAttribute((const void*)qkv_kernel,
                      hipFuncAttributeMaxDynamicSharedMemorySize, (int)sm1);
  hipFuncSetAttribute((const void*)attn_kernel,
                      hipFuncAttributeMaxDynamicSharedMemorySize, (int)sm2);
  hipFuncSetAttribute((const void*)stamp_kernel,
                      hipFuncAttributeMaxDynamicSharedMemorySize, (int)sm3);

  qkv_kernel<<<B_, 256, sm1, stream>>>(re, alia, Wq, Wk, Wv, wsQ, wsK, wsV);
  attn_kernel<<<B_, 256, sm2, stream>>>(re, alia, slen, wsQ, wsK, wsV, wsDec);
  stamp_kernel<<<B_, 256, sm3, stream>>>(slen, wsDec, u_w1, u_w2, u_b, u_v, out);
}